// MultiAdjNet_45767171506782
// MI455X (gfx1250) — compile-verified
//
#include <hip/hip_runtime.h>
#include <math.h>

typedef __attribute__((ext_vector_type(2))) float v2f;
typedef __attribute__((ext_vector_type(8))) float v8f;

#define N_NODES 100000
#define N_EDGES 600000
#define N_GRAPHS 256

// ---------------------------------------------------------------- utilities

__global__ void k_deg(float* __restrict__ deg, const int* __restrict__ idx, int E) {
    int t = blockIdx.x * blockDim.x + threadIdx.x;
    if (t < E) atomicAdd(&deg[idx[t]], 1.0f);
}

__global__ void k_dinv(float* __restrict__ d, int N) {
    int i = blockIdx.x * blockDim.x + threadIdx.x;
    if (i < N) {
        float v = d[i];
        d[i] = (v > 0.0f) ? rsqrtf(fmaxf(v, 1.0f)) : 0.0f;
    }
}

__global__ void k_edgew(float* __restrict__ w, const float* __restrict__ dinv,
                        const int* __restrict__ a, const int* __restrict__ b, int E) {
    int e = blockIdx.x * blockDim.x + threadIdx.x;
    if (e < E) w[e] = -dinv[a[e]] * dinv[b[e]];
}

__global__ void k_cnt(float* __restrict__ cnt, const int* __restrict__ batch, int N) {
    int i = blockIdx.x * blockDim.x + threadIdx.x;
    if (i < N) atomicAdd(&cnt[batch[i]], 1.0f);
}

__global__ void k_negcopy(float* __restrict__ out, const float* __restrict__ in, int n) {
    int i = blockIdx.x * blockDim.x + threadIdx.x;
    if (i < n) out[i] = -in[i];
}

// ------------------------------------------------- edge-parallel scatter SpMM
// out[sidx[e], :] += scale * w[e] * in[gidx[e], :]   (float4 per thread)

template<int F>
__global__ void k_spmm(float* __restrict__ out, const float* __restrict__ in,
                       const int* __restrict__ gidx, const int* __restrict__ sidx,
                       const float* __restrict__ w, float scale, int E) {
    constexpr int CH = F / 4;
    int t = blockIdx.x * blockDim.x + threadIdx.x;
    int e = t / CH;
    if (e >= E) return;
    int c = (t - e * CH) * 4;
    float ww = w[e] * scale;
    const float4 v = *(const float4*)(in + (size_t)gidx[e] * F + c);
    float* op = out + (size_t)sidx[e] * F + c;
    atomicAdd(op + 0, ww * v.x);
    atomicAdd(op + 1, ww * v.y);
    atomicAdd(op + 2, ww * v.z);
    atomicAdd(op + 3, ww * v.w);
}

// ------------------------------------------------------------- WMMA f32 GEMM
// C[16x16] per wave.  out = relu( sum_k A_k[M,F] @ W[k][F,NOUT] + bias ).
// Direct-store mode (batch == nullptr): out[row*out_stride + col_off + col].
// Pool mode: atomicAdd into out[batch[row]*out_stride + col_off + col].
//
// A 16x4 layout: lane<16 -> M=lane, K={0,1}; lane>=16 -> M=lane-16, K={2,3}
// B 4x16 layout (mirror): lane lo -> N=lo, hi selects K pair {2*hi, 2*hi+1}
// C layout: vgpr j, lane -> (M = j + 8*(lane>>4), N = lane&15)

template<int F, int NOUT>
__global__ void k_gemm_wmma(const float* __restrict__ Astack,   // [5][M][F]
                            const float* __restrict__ Wt,       // [5][F][NOUT]
                            const float* __restrict__ bias,     // [NOUT]
                            float* __restrict__ out,
                            const int* __restrict__ batch,
                            int M, int out_stride, int col_off, int ntiles) {
    int wave = threadIdx.x >> 5;
    int lane = threadIdx.x & 31;
    int tile = blockIdx.x * 4 + wave;
    if (tile >= ntiles) return;                 // uniform per-wave, EXEC stays full
    constexpr int TN = NOUT / 16;
    int tm = tile / TN, tn = tile - tm * TN;
    int m0 = tm * 16, n0 = tn * 16;
    int hi = lane >> 4, lo = lane & 15;
    size_t ss = (size_t)M * F;

    v8f acc = {0.f, 0.f, 0.f, 0.f, 0.f, 0.f, 0.f, 0.f};
    for (int k = 0; k < 5; ++k) {
        const float* A = Astack + (size_t)k * ss + (size_t)(m0 + lo) * F + 2 * hi;
        const float* B = Wt + (size_t)k * F * NOUT + (size_t)(2 * hi) * NOUT + n0 + lo;
#pragma unroll
        for (int c = 0; c < F; c += 4) {
            v2f a; a[0] = A[c];                 a[1] = A[c + 1];
            v2f b; b[0] = B[(size_t)c * NOUT];  b[1] = B[(size_t)(c + 1) * NOUT];
            acc = __builtin_amdgcn_wmma_f32_16x16x4_f32(
                false, a, false, b, (short)0, acc, false, false);
        }
    }

    float bv = bias[n0 + lo];
#pragma unroll
    for (int j = 0; j < 8; ++j) {
        int row = m0 + j + 8 * hi;
        float v = acc[j] + bv;
        v = v > 0.0f ? v : 0.0f;
        if (batch) {
            atomicAdd(&out[(size_t)batch[row] * out_stride + col_off + n0 + lo], v);
        } else {
            out[(size_t)row * out_stride + col_off + n0 + lo] = v;
        }
    }
}

// ------------------------------------------------------ pooled head + softmax

__global__ void k_head(float* __restrict__ out, const float* __restrict__ pool,
                       const float* __restrict__ cnt, const float* __restrict__ fcw,
                       const float* __restrict__ fcb, int G) {
    int g = blockIdx.x * blockDim.x + threadIdx.x;
    if (g >= G) return;
    float inv = 1.0f / fmaxf(cnt[g], 1.0f);
    const float* pr = pool + (size_t)g * 512;
    float z0 = 0.f, z1 = 0.f, z2 = 0.f, z3 = 0.f;
    for (int c = 0; c < 512; ++c) {
        float p = pr[c] * inv;
        const float* wr = fcw + c * 4;
        z0 += p * wr[0]; z1 += p * wr[1]; z2 += p * wr[2]; z3 += p * wr[3];
    }
    z0 += fcb[0]; z1 += fcb[1]; z2 += fcb[2]; z3 += fcb[3];
    float m = fmaxf(fmaxf(z0, z1), fmaxf(z2, z3));
    float s = expf(z0 - m) + expf(z1 - m) + expf(z2 - m) + expf(z3 - m);
    float lse = m + logf(s);
    float* o = out + (size_t)g * 4;
    o[0] = z0 - lse; o[1] = z1 - lse; o[2] = z2 - lse; o[3] = z3 - lse;
}

// --------------------------------------------------------------- host helpers

static inline unsigned nblk(long long n, int b) { return (unsigned)((n + b - 1) / b); }

// Build Chebyshev stack Tx[1..4] given Tx[0] already in slice 0.
template<int F>
static void build_chain(float* stack, int N, int E, const int* g, const int* s,
                        const float* w, hipStream_t stream) {
    size_t ss = (size_t)N * F;
    unsigned sb = nblk((long long)E * (F / 4), 256);
    unsigned eb = nblk((long long)ss, 256);
    hipMemsetAsync(stack + ss, 0, ss * sizeof(float), stream);
    k_spmm<F><<<sb, 256, 0, stream>>>(stack + ss, stack, g, s, w, 1.0f, E);
    for (int k = 2; k < 5; ++k) {
        k_negcopy<<<eb, 256, 0, stream>>>(stack + (size_t)k * ss,
                                          stack + (size_t)(k - 2) * ss, (int)ss);
        k_spmm<F><<<sb, 256, 0, stream>>>(stack + (size_t)k * ss,
                                          stack + (size_t)(k - 1) * ss, g, s, w, 2.0f, E);
    }
}

extern "C" void kernel_launch(void* const* d_in, const int* in_sizes, int n_in,
                              void* d_out, int out_size, void* d_ws, size_t ws_size,
                              hipStream_t stream) {
    const int N = N_NODES, E = N_EDGES, G = N_GRAPHS;

    const float* x    = (const float*)d_in[0];
    const int*   ei   = (const int*)d_in[1];
    const int*   src  = ei;
    const int*   dst  = ei + E;
    const int*   batch= (const int*)d_in[2];
    const float* W11  = (const float*)d_in[3];
    const float* b11  = (const float*)d_in[4];
    const float* W12  = (const float*)d_in[5];
    const float* b12  = (const float*)d_in[6];
    const float* W21  = (const float*)d_in[7];
    const float* b21  = (const float*)d_in[8];
    const float* W22  = (const float*)d_in[9];
    const float* b22  = (const float*)d_in[10];
    const float* fcw  = (const float*)d_in[11];
    const float* fcb  = (const float*)d_in[12];

    // workspace carve-up (floats)
    size_t need = ((size_t)2 * N + 2 * E + G + (size_t)G * 512 +
                   (size_t)N * 128 + (size_t)5 * N * 128) * sizeof(float);
    if (ws_size < need) return;

    float* ws      = (float*)d_ws;
    float* dinv_f  = ws;  ws += N;
    float* dinv_r  = ws;  ws += N;
    float* w_f     = ws;  ws += E;
    float* w_r     = ws;  ws += E;
    float* cnt     = ws;  ws += G;
    float* pool    = ws;  ws += (size_t)G * 512;
    float* h1      = ws;  ws += (size_t)N * 128;
    float* stack   = ws;  // 5 * N * 128 floats

    // --- normalization weights + pool counts ---------------------------------
    hipMemsetAsync(dinv_f, 0, sizeof(float) * 2 * N, stream);           // dinv_f|dinv_r
    hipMemsetAsync(cnt, 0, sizeof(float) * (G + (size_t)G * 512), stream); // cnt|pool
    k_deg<<<nblk(E, 256), 256, 0, stream>>>(dinv_f, src, E);
    k_deg<<<nblk(E, 256), 256, 0, stream>>>(dinv_r, dst, E);
    k_dinv<<<nblk(N, 256), 256, 0, stream>>>(dinv_f, N);
    k_dinv<<<nblk(N, 256), 256, 0, stream>>>(dinv_r, N);
    k_edgew<<<nblk(E, 256), 256, 0, stream>>>(w_f, dinv_f, src, dst, E);
    k_edgew<<<nblk(E, 256), 256, 0, stream>>>(w_r, dinv_r, src, dst, E);
    k_cnt<<<nblk(N, 256), 256, 0, stream>>>(cnt, batch, N);

    // --- layer 1: F=16 -> 64+64, write h1 [N,128] ----------------------------
    {
        const int tiles = (N / 16) * (64 / 16);   // 25000, divisible by 4
        hipMemcpyAsync(stack, x, sizeof(float) * (size_t)N * 16,
                       hipMemcpyDeviceToDevice, stream);
        build_chain<16>(stack, N, E, src, dst, w_f, stream);
        k_gemm_wmma<16, 64><<<tiles / 4, 128, 0, stream>>>(
            stack, W11, b11, h1, nullptr, N, 128, 0, tiles);
        build_chain<16>(stack, N, E, dst, src, w_r, stream);   // reversed adjacency
        k_gemm_wmma<16, 64><<<tiles / 4, 128, 0, stream>>>(
            stack, W12, b12, h1, nullptr, N, 128, 64, tiles);
    }

    // --- layer 2: F=128 -> 256+256, fused mean-pool accumulate ---------------
    {
        const int tiles = (N / 16) * (256 / 16);  // 100000, divisible by 4
        hipMemcpyAsync(stack, h1, sizeof(float) * (size_t)N * 128,
                       hipMemcpyDeviceToDevice, stream);
        build_chain<128>(stack, N, E, src, dst, w_f, stream);
        k_gemm_wmma<128, 256><<<tiles / 4, 128, 0, stream>>>(
            stack, W21, b21, pool, batch, N, 512, 0, tiles);
        build_chain<128>(stack, N, E, dst, src, w_r, stream);
        k_gemm_wmma<128, 256><<<tiles / 4, 128, 0, stream>>>(
            stack, W22, b22, pool, batch, N, 512, 256, tiles);
    }

    // --- head: divide by counts, FC 512->4, log_softmax ----------------------
    k_head<<<1, G, 0, stream>>>((float*)d_out, pool, cnt, fcw, fcb, G);
}